// AttnMDN_62629213110805
// MI455X (gfx1250) — compile-verified
//
#include <hip/hip_runtime.h>

#define Nn 100000
#define Ee 1600000
#define FIN 30
#define XSTR 64   // padded row stride for xw / out buffers

typedef float v2f __attribute__((ext_vector_type(2)));
typedef float v8f __attribute__((ext_vector_type(8)));

// ---------- helpers ----------
__device__ __forceinline__ void atomicMaxF(float* addr, float val) {
    // int ordering == float ordering for non-negative; reversed unsigned for negative
    if (val >= 0.0f) atomicMax((int*)addr, __float_as_int(val));
    else             atomicMin((unsigned int*)addr, __float_as_uint(val));
}
__device__ __forceinline__ float leaky(float x) { return x > 0.0f ? x : 0.2f * x; }

// ---------- K0: zero scratch header + ew accumulators ----------
__global__ void k_zero(float* p, int count) {
    int i = blockIdx.x * blockDim.x + threadIdx.x;
    if (i < count) p[i] = 0.0f;
}

// ---------- K1: BatchNorm column stats ----------
__global__ void k_bn_stats(const float* __restrict__ h, float* colsum, float* colsumsq) {
    __shared__ float ls[FIN], lq[FIN];
    int t = threadIdx.x;
    if (t < FIN) { ls[t] = 0.0f; lq[t] = 0.0f; }
    __syncthreads();
    int n = blockIdx.x * blockDim.x + t;
    if (n < Nn) {
        const float* row = h + (size_t)n * FIN;
        #pragma unroll
        for (int k = 0; k < FIN; ++k) {
            float v = row[k];
            atomicAdd(&ls[k], v);
            atomicAdd(&lq[k], v * v);
        }
    }
    __syncthreads();
    if (t < FIN) { atomicAdd(&colsum[t], ls[t]); atomicAdd(&colsumsq[t], lq[t]); }
}

// ---------- K2: per-dst edge-weight sum/count (for self-loop fill 'mean') ----------
__global__ void k_ew_acc(const int* __restrict__ dst, const float* __restrict__ ew,
                         float* ew_sum, float* ew_cnt) {
    int e = blockIdx.x * blockDim.x + threadIdx.x;
    if (e >= Ee) return;
    int d = dst[e];
    atomicAdd(&ew_sum[d], ew[e]);
    atomicAdd(&ew_cnt[d], 1.0f);
}

// ---------- K3: finalize BN scale/shift (padded to 32) + ke[h] ----------
__global__ void k_finalize(const float* colsum, const float* colsumsq,
                           const float* __restrict__ gamma, const float* __restrict__ beta,
                           const float* __restrict__ W_edge, const float* __restrict__ att_edge,
                           float* scalev, float* shiftv, float* ke) {
    int t = threadIdx.x;
    if (t < 32) {
        float s = 0.0f, sh = 0.0f;
        if (t < FIN) {
            float mu  = colsum[t]   * (1.0f / (float)Nn);
            float var = colsumsq[t] * (1.0f / (float)Nn) - mu * mu;
            float rs  = rsqrtf(var + 1e-5f);
            s  = gamma[t] * rs;
            sh = beta[t] - mu * s;
        }
        scalev[t] = s;
        shiftv[t] = sh;
    }
    if (t < 2) {
        float acc = 0.0f;
        #pragma unroll
        for (int c = 0; c < 30; ++c) acc += W_edge[t * 30 + c] * att_edge[t * 30 + c];
        ke[t] = acc;
    }
}

// ---------- K4: fused BN-normalize + GEMM (x @ W_lin) via V_WMMA_F32_16X16X4_F32 ----------
// Each wave computes a 16x64 tile of xw; 4 col-tiles x 8 k-steps = 32 wmma/wave.
__global__ void k_gemm_xw(const float* __restrict__ h, const float* __restrict__ Wlin,
                          const float* __restrict__ scalev, const float* __restrict__ shiftv,
                          float* __restrict__ xw) {
    __shared__ float ldsW[32][XSTR];   // K padded 30->32, N padded 60->64
    int t = threadIdx.x;
    for (int i = t; i < 32 * XSTR; i += blockDim.x) {
        int k = i >> 6, n = i & 63;
        ldsW[k][n] = (k < FIN && n < 60) ? Wlin[k * 60 + n] : 0.0f;
    }
    __syncthreads();

    int wave   = t >> 5;
    int lane   = t & 31;
    int laneLo = lane & 15;
    int laneHi = lane >> 4;
    int tile   = blockIdx.x * 4 + wave;          // 16-row tile; Nn/16 = 6250 exactly
    if (tile >= Nn / 16) return;                 // wave-uniform exit: EXEC stays all-1 for WMMA

    int row = tile * 16 + laneLo;
    const float* hrow = h + (size_t)row * FIN;

    v8f acc0 = {}, acc1 = {}, acc2 = {}, acc3 = {};
    #pragma unroll
    for (int kk = 0; kk < 8; ++kk) {
        int k0 = kk * 4 + 2 * laneHi;            // A layout: lanes0-15 hold K=k0..k0+1, lanes16-31 K=k0+2..k0+3
        v2f a;
        a.x = (k0     < FIN) ? hrow[k0]     * scalev[k0]     + shiftv[k0]     : 0.0f;
        a.y = (k0 + 1 < FIN) ? hrow[k0 + 1] * scalev[k0 + 1] + shiftv[k0 + 1] : 0.0f;

        v2f b0, b1, b2, b3;
        b0.x = ldsW[k0][laneLo +  0]; b0.y = ldsW[k0 + 1][laneLo +  0];
        b1.x = ldsW[k0][laneLo + 16]; b1.y = ldsW[k0 + 1][laneLo + 16];
        b2.x = ldsW[k0][laneLo + 32]; b2.y = ldsW[k0 + 1][laneLo + 32];
        b3.x = ldsW[k0][laneLo + 48]; b3.y = ldsW[k0 + 1][laneLo + 48];

        acc0 = __builtin_amdgcn_wmma_f32_16x16x4_f32(false, a, false, b0, (short)0, acc0, false, false);
        acc1 = __builtin_amdgcn_wmma_f32_16x16x4_f32(false, a, false, b1, (short)0, acc1, false, false);
        acc2 = __builtin_amdgcn_wmma_f32_16x16x4_f32(false, a, false, b2, (short)0, acc2, false, false);
        acc3 = __builtin_amdgcn_wmma_f32_16x16x4_f32(false, a, false, b3, (short)0, acc3, false, false);
    }

    // D layout: VGPR v -> row = v + 8*laneHi, col = laneLo (per 16x16 tile)
    #pragma unroll
    for (int v = 0; v < 8; ++v) {
        int r = tile * 16 + v + 8 * laneHi;
        float* dstp = xw + (size_t)r * XSTR;
        dstp[laneLo +  0] = acc0[v];
        dstp[laneLo + 16] = acc1[v];
        dstp[laneLo + 32] = acc2[v];
        dstp[laneLo + 48] = acc3[v];
    }
}

// ---------- K5: per-node attention terms + self-loop alpha (init of segment max) ----------
__global__ void k_node_attn(const float* __restrict__ xw,
                            const float* __restrict__ att_src, const float* __restrict__ att_dst,
                            const float* __restrict__ ke,
                            const float* __restrict__ ew_sum, const float* __restrict__ ew_cnt,
                            float* a_srcv, float* a_dstv, float* alphaL, float* mbuf) {
    int n = blockIdx.x * blockDim.x + threadIdx.x;
    if (n >= Nn) return;
    const float* xr = xw + (size_t)n * XSTR;
    float la = ew_sum[n] / fmaxf(ew_cnt[n], 1.0f);
    #pragma unroll
    for (int hh = 0; hh < 2; ++hh) {
        float as = 0.0f, ad = 0.0f;
        #pragma unroll
        for (int c = 0; c < 30; ++c) {
            float v = xr[hh * 30 + c];
            as += v * att_src[hh * 30 + c];
            ad += v * att_dst[hh * 30 + c];
        }
        a_srcv[n * 2 + hh] = as;
        a_dstv[n * 2 + hh] = ad;
        float al = leaky(as + ad + ke[hh] * la);
        alphaL[n * 2 + hh] = al;
        mbuf[n * 2 + hh]   = al;   // self-loop seeds the segment max
    }
}

// ---------- K6: edge alpha + segment max ----------
__global__ void k_edge_alpha(const int* __restrict__ src, const int* __restrict__ dst,
                             const float* __restrict__ ew,
                             const float* __restrict__ a_srcv, const float* __restrict__ a_dstv,
                             const float* __restrict__ ke,
                             float* edgebuf, float* mbuf) {
    int tid = blockIdx.x * blockDim.x + threadIdx.x;
    if (tid >= 2 * Ee) return;
    int e = tid >> 1, hh = tid & 1;
    int s = src[e], d = dst[e];
    float al = leaky(a_srcv[s * 2 + hh] + a_dstv[d * 2 + hh] + ke[hh] * ew[e]);
    edgebuf[(size_t)e * 2 + hh] = al;
    atomicMaxF(&mbuf[d * 2 + hh], al);
}

// ---------- K7: denominator init with self-loop term ----------
__global__ void k_den_init(const float* __restrict__ alphaL, const float* __restrict__ mbuf,
                           float* den) {
    int n = blockIdx.x * blockDim.x + threadIdx.x;
    if (n >= Nn) return;
    den[n * 2 + 0] = expf(alphaL[n * 2 + 0] - mbuf[n * 2 + 0]);
    den[n * 2 + 1] = expf(alphaL[n * 2 + 1] - mbuf[n * 2 + 1]);
}

// ---------- K8: edge exp + segment sum ----------
__global__ void k_edge_exp(const int* __restrict__ dst, const float* __restrict__ mbuf,
                           float* edgebuf, float* den) {
    int tid = blockIdx.x * blockDim.x + threadIdx.x;
    if (tid >= 2 * Ee) return;
    int e = tid >> 1, hh = tid & 1;
    int d = dst[e];
    float ex = expf(edgebuf[(size_t)e * 2 + hh] - mbuf[d * 2 + hh]);
    edgebuf[(size_t)e * 2 + hh] = ex;
    atomicAdd(&den[d * 2 + hh], ex);
}

// ---------- K9: out init with self-loop message ----------
__global__ void k_out_init(const float* __restrict__ alphaL, const float* __restrict__ mbuf,
                           const float* __restrict__ den, const float* __restrict__ xw,
                           float* outb) {
    int n = blockIdx.x * blockDim.x + threadIdx.x;
    if (n >= Nn) return;
    #pragma unroll
    for (int hh = 0; hh < 2; ++hh) {
        float p = expf(alphaL[n * 2 + hh] - mbuf[n * 2 + hh]) / den[n * 2 + hh];
        const float* xs = xw   + (size_t)n * XSTR + hh * 30;
        float*       od = outb + (size_t)n * XSTR + hh * 30;
        #pragma unroll
        for (int c = 0; c < 30; ++c) od[c] = p * xs[c];
    }
}

// ---------- K10: edge message scatter (dominant pass) ----------
__global__ void k_edge_scatter(const int* __restrict__ src, const int* __restrict__ dst,
                               const float* __restrict__ edgebuf, const float* __restrict__ den,
                               const float* __restrict__ xw, float* outb) {
    int tid = blockIdx.x * blockDim.x + threadIdx.x;
    if (tid >= 2 * Ee) return;
    int e = tid >> 1, hh = tid & 1;
    int s = src[e], d = dst[e];
    float p = edgebuf[(size_t)e * 2 + hh] / den[d * 2 + hh];
    const float* xs = xw   + (size_t)s * XSTR + hh * 30;
    float*       od = outb + (size_t)d * XSTR + hh * 30;
    #pragma unroll
    for (int c = 0; c < 30; ++c) atomicAdd(&od[c], p * xs[c]);
}

// ---------- K11: MDN MLP head ----------
__global__ void k_mlp(const float* __restrict__ outb, const float* __restrict__ bias_conv,
                      const float* __restrict__ fc1w, const float* __restrict__ fc1b,
                      const float* __restrict__ fc2w, const float* __restrict__ fc2b,
                      const float* __restrict__ fc3w, const float* __restrict__ fc3b,
                      const float* __restrict__ fc4w, const float* __restrict__ fc4b,
                      const float* __restrict__ fc5w, const float* __restrict__ fc5b,
                      float* __restrict__ out) {
    __shared__ float sw1[600], sw2[100], sw3[100];
    __shared__ float sb1[10], sb2[10], sb3[10], sw4[10], sw5[10], sbc[60];
    __shared__ float sb4, sb5;
    int t = threadIdx.x;
    for (int i = t; i < 600; i += blockDim.x) sw1[i] = fc1w[i];
    for (int i = t; i < 100; i += blockDim.x) { sw2[i] = fc2w[i]; sw3[i] = fc3w[i]; }
    for (int i = t; i < 60;  i += blockDim.x) sbc[i] = bias_conv[i];
    if (t < 10) { sb1[t] = fc1b[t]; sb2[t] = fc2b[t]; sb3[t] = fc3b[t]; sw4[t] = fc4w[t]; sw5[t] = fc5w[t]; }
    if (t == 0) { sb4 = fc4b[0]; sb5 = fc5b[0]; }
    __syncthreads();

    int n = blockIdx.x * blockDim.x + t;
    if (n >= Nn) return;
    const float* row = outb + (size_t)n * XSTR;

    float z1[10];
    #pragma unroll
    for (int j = 0; j < 10; ++j) z1[j] = sb1[j];
    for (int i = 0; i < 60; ++i) {
        float v = row[i] + sbc[i];
        v = v > 0.0f ? v : 0.0f;              // relu(hgat)
        #pragma unroll
        for (int j = 0; j < 10; ++j) z1[j] += v * sw1[i * 10 + j];
    }
    #pragma unroll
    for (int j = 0; j < 10; ++j) z1[j] = z1[j] > 0.0f ? z1[j] : 0.0f;

    float z2[10];
    #pragma unroll
    for (int j = 0; j < 10; ++j) {
        float acc = sb2[j];
        #pragma unroll
        for (int i = 0; i < 10; ++i) acc += z1[i] * sw2[i * 10 + j];
        z2[j] = acc > 0.0f ? acc : 0.0f;
    }
    float z3[10];
    #pragma unroll
    for (int j = 0; j < 10; ++j) {
        float acc = sb3[j];
        #pragma unroll
        for (int i = 0; i < 10; ++i) acc += z2[i] * sw3[i * 10 + j];
        z3[j] = acc;
    }
    float va = sb4, vb = sb5;
    #pragma unroll
    for (int i = 0; i < 10; ++i) { va += z3[i] * sw4[i]; vb += z3[i] * sw5[i]; }
    float a = (va > 0.0f ? va : expf(va) - 1.0f) + 1.0f;
    float b = (vb > 0.0f ? vb : expf(vb) - 1.0f) + 1.0f;
    out[n] = a;
    out[Nn + n] = b;
}

// ---------- launcher ----------
extern "C" void kernel_launch(void* const* d_in, const int* in_sizes, int n_in,
                              void* d_out, int out_size, void* d_ws, size_t ws_size,
                              hipStream_t stream) {
    const float* h         = (const float*)d_in[0];
    const int*   ei        = (const int*)  d_in[1];
    const float* ew        = (const float*)d_in[2];
    const float* gamma     = (const float*)d_in[3];
    const float* beta      = (const float*)d_in[4];
    const float* Wlin      = (const float*)d_in[5];
    const float* att_src   = (const float*)d_in[6];
    const float* att_dst   = (const float*)d_in[7];
    const float* W_edge    = (const float*)d_in[8];
    const float* att_edge  = (const float*)d_in[9];
    const float* bias_conv = (const float*)d_in[10];
    const float* fc1w = (const float*)d_in[11]; const float* fc1b = (const float*)d_in[12];
    const float* fc2w = (const float*)d_in[13]; const float* fc2b = (const float*)d_in[14];
    const float* fc3w = (const float*)d_in[15]; const float* fc3b = (const float*)d_in[16];
    const float* fc4w = (const float*)d_in[17]; const float* fc4b = (const float*)d_in[18];
    const float* fc5w = (const float*)d_in[19]; const float* fc5b = (const float*)d_in[20];
    float* out = (float*)d_out;

    const int* srcp = ei;        // edge_index[0]
    const int* dstp = ei + Ee;   // edge_index[1]

    // workspace layout (floats)
    float* ws       = (float*)d_ws;
    float* colsum   = ws + 0;                    // 32
    float* colsumsq = ws + 32;                   // 32
    float* scalev   = ws + 64;                   // 32
    float* shiftv   = ws + 96;                   // 32
    float* ke       = ws + 128;                  // 2 (pad to 32)
    float* ew_sum   = ws + 160;                  // N
    float* ew_cnt   = ew_sum + Nn;               // N
    float* a_srcv   = ew_cnt + Nn;               // 2N
    float* a_dstv   = a_srcv + 2 * Nn;           // 2N
    float* alphaL   = a_dstv + 2 * Nn;           // 2N
    float* mbuf     = alphaL + 2 * Nn;           // 2N
    float* den      = mbuf   + 2 * Nn;           // 2N
    float* xw       = den    + 2 * Nn;           // 64N
    float* outb     = xw     + (size_t)XSTR * Nn;// 64N
    float* edgebuf  = outb   + (size_t)XSTR * Nn;// 2E

    const int zeroCnt = 160 + 2 * Nn;
    k_zero<<<(zeroCnt + 255) / 256, 256, 0, stream>>>(ws, zeroCnt);
    k_bn_stats<<<(Nn + 255) / 256, 256, 0, stream>>>(h, colsum, colsumsq);
    k_ew_acc<<<(Ee + 255) / 256, 256, 0, stream>>>(dstp, ew, ew_sum, ew_cnt);
    k_finalize<<<1, 64, 0, stream>>>(colsum, colsumsq, gamma, beta, W_edge, att_edge,
                                     scalev, shiftv, ke);
    k_gemm_xw<<<(Nn / 16 + 3) / 4, 128, 0, stream>>>(h, Wlin, scalev, shiftv, xw);
    k_node_attn<<<(Nn + 255) / 256, 256, 0, stream>>>(xw, att_src, att_dst, ke,
                                                      ew_sum, ew_cnt,
                                                      a_srcv, a_dstv, alphaL, mbuf);
    k_edge_alpha<<<(2 * Ee + 255) / 256, 256, 0, stream>>>(srcp, dstp, ew, a_srcv, a_dstv,
                                                           ke, edgebuf, mbuf);
    k_den_init<<<(Nn + 255) / 256, 256, 0, stream>>>(alphaL, mbuf, den);
    k_edge_exp<<<(2 * Ee + 255) / 256, 256, 0, stream>>>(dstp, mbuf, edgebuf, den);
    k_out_init<<<(Nn + 255) / 256, 256, 0, stream>>>(alphaL, mbuf, den, xw, outb);
    k_edge_scatter<<<(2 * Ee + 255) / 256, 256, 0, stream>>>(srcp, dstp, edgebuf, den,
                                                             xw, outb);
    k_mlp<<<(Nn + 255) / 256, 256, 0, stream>>>(outb, bias_conv,
                                                fc1w, fc1b, fc2w, fc2b, fc3w, fc3b,
                                                fc4w, fc4b, fc5w, fc5b, out);
}